// NeRFLoss_85779086835715
// MI455X (gfx1250) — compile-verified
//
#include <hip/hip_runtime.h>

#define NRAYS 65536
#define SPR 128                 // samples per ray
#define WPB 8                   // waves (rays) per block
#define LAMBDA_OPACITY 0.001f
#define LAMBDA_DISTORTION 0.001f

// ---------------------------------------------------------------------------
// Single fused kernel:
//  * every block: 8 rays of distortion loss (one wave32 per ray), streamed
//    through the gfx1250 async global->LDS DMA path (ASYNCcnt-tracked)
//  * blocks 0..255: additionally compute the elementwise d_rgb / d_opacity
//    slice, scheduled while the async DMAs are in flight so it is free
// ---------------------------------------------------------------------------
__global__ __launch_bounds__(256) void nerf_loss_fused_kernel(
    const float* __restrict__ ws, const float* __restrict__ deltas,
    const float* __restrict__ ts, const long long* __restrict__ rays_a,
    const float4* __restrict__ rgb_pred, const float4* __restrict__ rgb_target,
    const float4* __restrict__ opacity, float4* __restrict__ out_rgb,
    float4* __restrict__ out_op, float* __restrict__ out_dist) {
  __shared__ float s_w[WPB * SPR];
  __shared__ float s_d[WPB * SPR];
  __shared__ float s_t[WPB * SPR];

  const int lane = threadIdx.x & 31;
  const int wv = threadIdx.x >> 5;
  const int ray = blockIdx.x * WPB + wv;

  const long long ridx = rays_a[3 * ray + 0];
  const int start = (int)rays_a[3 * ray + 1];  // < 2^23, fits in int
  const int gbase = start + lane * 4;
  const int lidx = wv * SPR + lane * 4;

  // gfx1250 async DMA: each lane pulls its 16B of each stream into LDS.
  // Flat LDS addr bits [31:0] == workgroup-relative LDS byte address.
  unsigned lw = (unsigned)(size_t)(&s_w[lidx]);
  unsigned ld = (unsigned)(size_t)(&s_d[lidx]);
  unsigned lt = (unsigned)(size_t)(&s_t[lidx]);
  const float* gw = ws + gbase;
  const float* gd = deltas + gbase;
  const float* gt = ts + gbase;
  asm volatile("global_load_async_to_lds_b128 %0, %1, off" ::"v"(lw), "v"(gw)
               : "memory");
  asm volatile("global_load_async_to_lds_b128 %0, %1, off" ::"v"(ld), "v"(gd)
               : "memory");
  asm volatile("global_load_async_to_lds_b128 %0, %1, off" ::"v"(lt), "v"(gt)
               : "memory");

  // ---- elementwise slice, hidden under the async DMA latency --------------
  {
    constexpr int NRGB4 = NRAYS * 3 / 4;  // 49152
    constexpr int NOP4 = NRAYS / 4;       // 16384
    int gi = blockIdx.x * 256 + threadIdx.x;  // only blocks 0..255 hit
    if (gi < NRGB4) {
      float4 a = rgb_pred[gi], b = rgb_target[gi], r;
      r.x = (a.x - b.x) * (a.x - b.x) + 1e-5f;
      r.y = (a.y - b.y) * (a.y - b.y) + 1e-5f;
      r.z = (a.z - b.z) * (a.z - b.z) + 1e-5f;
      r.w = (a.w - b.w) * (a.w - b.w) + 1e-5f;
      out_rgb[gi] = r;
    } else if (gi < NRGB4 + NOP4) {
      int j = gi - NRGB4;
      float4 o = opacity[j], r;
      float ox = o.x + 1e-5f, oy = o.y + 1e-5f;
      float oz = o.z + 1e-5f, ow = o.w + 1e-5f;
      r.x = -LAMBDA_OPACITY * ox * __logf(ox);
      r.y = -LAMBDA_OPACITY * oy * __logf(oy);
      r.z = -LAMBDA_OPACITY * oz * __logf(oz);
      r.w = -LAMBDA_OPACITY * ow * __logf(ow);
      out_op[j] = r;
    }
  }

  // ---- wait for DMA, then scan ------------------------------------------
  asm volatile("s_wait_asynccnt 0" ::: "memory");

  const float4 w4 = *(const float4*)(&s_w[lidx]);
  const float4 d4 = *(const float4*)(&s_d[lidx]);
  const float4 t4 = *(const float4*)(&s_t[lidx]);

  const float wt0 = w4.x * t4.x, wt1 = w4.y * t4.y;
  const float wt2 = w4.z * t4.z, wt3 = w4.w * t4.w;

  // lane-local inclusive scans (4 elements each)
  const float sw0 = w4.x, sw1 = sw0 + w4.y, sw2 = sw1 + w4.z, sw3 = sw2 + w4.w;
  const float st0 = wt0, st1 = st0 + wt1, st2 = st1 + wt2, st3 = st2 + wt3;

  // wave32 inclusive scan of lane totals (two accumulators through 5 steps)
  float incW = sw3, incT = st3;
#pragma unroll
  for (int off = 1; off < 32; off <<= 1) {
    float aw = __shfl_up(incW, off, 32);
    float at = __shfl_up(incT, off, 32);
    if (lane >= off) {
      incW += aw;
      incT += at;
    }
  }
  const float excW = incW - sw3;  // exclusive prefix of sum(w)
  const float excT = incT - st3;  // exclusive prefix of sum(w*t)

  // contrib_j = 2*w_j*(t_j*S_j - T_j) + w_j^2 * delta_j / 3
  const float k3 = 1.0f / 3.0f;
  float c;
  c = 2.0f * w4.x * (t4.x * (excW + sw0) - (excT + st0)) + w4.x * w4.x * d4.x * k3;
  c += 2.0f * w4.y * (t4.y * (excW + sw1) - (excT + st1)) + w4.y * w4.y * d4.y * k3;
  c += 2.0f * w4.z * (t4.z * (excW + sw2) - (excT + st2)) + w4.z * w4.z * d4.z * k3;
  c += 2.0f * w4.w * (t4.w * (excW + sw3) - (excT + st3)) + w4.w * w4.w * d4.w * k3;

  // wave32 butterfly reduction
#pragma unroll
  for (int off = 16; off >= 1; off >>= 1) c += __shfl_xor(c, off, 32);

  if (lane == 0) out_dist[(int)ridx] = LAMBDA_DISTORTION * c;
}

// ---------------------------------------------------------------------------
extern "C" void kernel_launch(void* const* d_in, const int* in_sizes, int n_in,
                              void* d_out, int out_size, void* d_ws,
                              size_t ws_size, hipStream_t stream) {
  const float* rgb_pred = (const float*)d_in[0];
  const float* rgb_target = (const float*)d_in[1];
  const float* opacity = (const float*)d_in[2];
  const float* ws = (const float*)d_in[3];
  const float* deltas = (const float*)d_in[4];
  const float* ts = (const float*)d_in[5];
  const long long* rays_a = (const long long*)d_in[6];

  float* out = (float*)d_out;
  float* out_rgb = out;               // 196608 floats
  float* out_op = out + NRAYS * 3;    // 65536 floats (byte offset 786432, 16B aligned)
  float* out_dist = out + NRAYS * 4;  // 65536 floats

  // Single launch: 8192 blocks x 256 threads (one wave32 per ray);
  // elementwise slice folded into blocks 0..255 under the DMA shadow.
  nerf_loss_fused_kernel<<<NRAYS / WPB, 256, 0, stream>>>(
      ws, deltas, ts, rays_a, (const float4*)rgb_pred,
      (const float4*)rgb_target, (const float4*)opacity, (float4*)out_rgb,
      (float4*)out_op, out_dist);
}